// SamePadConvV3_34368328302981
// MI455X (gfx1250) — compile-verified
//
#include <hip/hip_runtime.h>
#include <hip/hip_bf16.h>

typedef __attribute__((ext_vector_type(2))) float v2f;
typedef __attribute__((ext_vector_type(8))) float v8f;
typedef __attribute__((ext_vector_type(4))) unsigned v4u;
typedef __attribute__((ext_vector_type(8))) unsigned v8u;

#define N_BATCH   64
#define C_OUT     64
#define C_IN      64
#define KW        3
#define LLEN      8192
#define CALIB     2.0f

// workspace layout (floats)
#define WS_Q      0        // 4416
#define WS_WMOD   4416     // 12288  (co, ci, k) like conv_weight
#define WS_BEFF   16704    // 64
#define WS_FEFF   16768    // 64

__device__ __forceinline__ float sigmoidf_(float x) { return 1.0f / (1.0f + __expf(-x)); }
__device__ __forceinline__ float siluf_(float x)    { return x * sigmoidf_(x); }

// ---------------------------------------------------------------------------
// Kernel 1: controller / hypernetwork. One workgroup, 256 threads.
// Produces wmod, bias_eff, f_eff in workspace.
// ---------------------------------------------------------------------------
__global__ __launch_bounds__(256) void controller_kernel(
    const float* __restrict__ conv_weight, const float* __restrict__ bias,
    const float* __restrict__ grads,  const float* __restrict__ q_ema,
    const float* __restrict__ ctrl_w, const float* __restrict__ ctrl_b,
    const float* __restrict__ gp_w,   const float* __restrict__ gp_b,
    const float* __restrict__ gt_w,   const float* __restrict__ gt_b,
    const float* __restrict__ ln_g,   const float* __restrict__ ln_b,
    const float* __restrict__ ccin_w, const float* __restrict__ ccin_b,
    const float* __restrict__ ck_w,   const float* __restrict__ ck_b,
    const float* __restrict__ cb_w,   const float* __restrict__ cb_b,
    const float* __restrict__ cf_w,   const float* __restrict__ cf_b,
    const float* __restrict__ tau_w,  const float* __restrict__ tau_b,
    const float* __restrict__ mem_W,  const float* __restrict__ temp_logit,
    float* __restrict__ ws)
{
    __shared__ float sh_h[64 * 64];
    __shared__ float sh_hm[64 * 64];
    __shared__ float sh_gn[64];
    __shared__ float sh_att[128];
    __shared__ float sh_red[64];
    __shared__ float sh_tau;
    __shared__ float sh_topv[4][3];
    __shared__ int   sh_topi[4][3];

    const int t = threadIdx.x;
    float* qbuf = ws + WS_Q;
    float* wmod = ws + WS_WMOD;
    float* beff = ws + WS_BEFF;
    float* feff = ws + WS_FEFF;

    // ---- stage 1: h = silu(g @ ctrl_w.T + ctrl_b); gn = ||g|| -------------
    for (int idx = t; idx < 4096; idx += 256) {
        const int i = idx >> 6, o = idx & 63;
        const float* gi = grads  + i * 192;
        const float* wo = ctrl_w + o * 192;
        float s = ctrl_b[o];
        for (int d = 0; d < 192; ++d) s += gi[d] * wo[d];
        sh_h[idx] = siluf_(s);
    }
    if (t < 64) {
        const float* gi = grads + t * 192;
        float s = 0.0f;
        for (int d = 0; d < 192; ++d) s += gi[d] * gi[d];
        sh_gn[t] = sqrtf(s);
    }
    __syncthreads();

    // ---- stage 2: gated mix hm = (1-z)h + z*silu(h@gt_w.T) ----------------
    for (int idx = t; idx < 4096; idx += 256) {
        const int i = idx >> 6, o = idx & 63;
        const float* hi = sh_h + i * 64;
        float zs = gp_b[o];
        const float* gpo = gp_w + o * 65;
        for (int d = 0; d < 64; ++d) zs += hi[d] * gpo[d];
        zs += sh_gn[i] * gpo[64];
        const float z = sigmoidf_(zs);
        float hs = gt_b[o];
        const float* gto = gt_w + o * 64;
        for (int d = 0; d < 64; ++d) hs += hi[d] * gto[d];
        sh_hm[idx] = (1.0f - z) * sh_h[idx] + z * siluf_(hs);
    }
    __syncthreads();

    // ---- stage 3: LayerNorm rows ------------------------------------------
    if (t < 64) {
        const float* r = sh_hm + t * 64;
        float mu = 0.0f;
        for (int d = 0; d < 64; ++d) mu += r[d];
        mu *= (1.0f / 64.0f);
        float var = 0.0f;
        for (int d = 0; d < 64; ++d) { float u = r[d] - mu; var += u * u; }
        var *= (1.0f / 64.0f);
        const float rstd = rsqrtf(var + 1e-5f);
        for (int d = 0; d < 64; ++d)
            sh_h[t * 64 + d] = ln_g[d] * (r[d] - mu) * rstd + ln_b[d];
    }
    __syncthreads();

    // ---- stage 4: q = 0.7*(calib heads) + 0.3*q_ema; tau ------------------
    for (int idx = t; idx < 4096; idx += 256) {
        const int i = idx >> 6, o = idx & 63;
        const float* hi = sh_h + i * 64;
        float s = ccin_b[o];
        const float* wo = ccin_w + o * 64;
        for (int d = 0; d < 64; ++d) s += hi[d] * wo[d];
        qbuf[idx] = 0.7f * (tanhf(s) * CALIB) + 0.3f * q_ema[idx];
    }
    for (int idx = t; idx < 192; idx += 256) {
        const int i = idx / 3, k = idx % 3;
        const float* hi = sh_h + i * 64;
        float s = ck_b[k];
        const float* wo = ck_w + k * 64;
        for (int d = 0; d < 64; ++d) s += hi[d] * wo[d];
        qbuf[4096 + idx] = 0.7f * (tanhf(s) * CALIB) + 0.3f * q_ema[4096 + idx];
    }
    if (t < 64) {
        const float* hi = sh_h + t * 64;
        float sb = cb_b[0], sf = cf_b[0], st = tau_b[0];
        for (int d = 0; d < 64; ++d) {
            sb += hi[d] * cb_w[d];
            sf += hi[d] * cf_w[d];
            st += hi[d] * tau_w[d];
        }
        qbuf[4288 + t] = 0.7f * (tanhf(sb) * CALIB) + 0.3f * q_ema[4288 + t];
        qbuf[4352 + t] = 0.7f * (tanhf(sf) * CALIB) + 0.3f * q_ema[4352 + t];
        sh_red[t] = sigmoidf_(st);
    }
    __syncthreads();
    if (t == 0) {
        float s = 0.0f;
        for (int d = 0; d < 64; ++d) s += sh_red[d];
        sh_tau = (s * (1.0f / 64.0f)) * 0.5f + 0.5f;
    }
    __syncthreads();

    // ---- stage 5: att[g][m] = q_groups[g] . mem_W[g][:][m] ----------------
    if (t < 128) {
        const int g = t >> 5, m = t & 31;
        const float* Wg = mem_W + (size_t)g * 1104 * 32 + m;
        float s = 0.0f;
        for (int d = 0; d < 1024; ++d) s += qbuf[g * 1024 + d] * Wg[d * 32];
        for (int d = 0; d < 48;  ++d) s += qbuf[4096 + g * 48 + d] * Wg[(1024 + d) * 32];
        for (int d = 0; d < 16;  ++d) s += qbuf[4288 + g * 16 + d] * Wg[(1072 + d) * 32];
        for (int d = 0; d < 16;  ++d) s += qbuf[4352 + g * 16 + d] * Wg[(1088 + d) * 32];
        sh_att[t] = s;
    }
    __syncthreads();

    // ---- stage 6: softmax(att/temp) + top-3 per group ---------------------
    if (t < 4) {
        const float temp = sigmoidf_(temp_logit[t]) * 0.6f + 0.2f;
        const float invt = 1.0f / (temp + 1e-8f);
        float mx = -1e30f;
        for (int m = 0; m < 32; ++m) mx = fmaxf(mx, sh_att[t * 32 + m] * invt);
        float p[32], den = 0.0f;
        for (int m = 0; m < 32; ++m) { p[m] = __expf(sh_att[t * 32 + m] * invt - mx); den += p[m]; }
        const float inv = 1.0f / den;
        for (int tt = 0; tt < 3; ++tt) {
            float bv = -1.0f; int bi = 0;
            for (int m = 0; m < 32; ++m) if (p[m] > bv) { bv = p[m]; bi = m; }
            sh_topv[t][tt] = bv * inv;
            sh_topi[t][tt] = bi;
            p[bi] = -2.0f;
        }
    }
    __syncthreads();

    // ---- stage 7: q = tau*q + (1-tau)*retrieved ---------------------------
    const float tau = sh_tau;
    for (int r = t; r < 4416; r += 256) {
        int g, d;
        if (r < 4096)      { g = r >> 10; d = r & 1023; }
        else if (r < 4288) { const int u = r - 4096; g = u / 48; d = 1024 + u % 48; }
        else if (r < 4352) { const int u = r - 4288; g = u >> 4; d = 1072 + (u & 15); }
        else               { const int u = r - 4352; g = u >> 4; d = 1088 + (u & 15); }
        const float* Wr = mem_W + ((size_t)g * 1104 + d) * 32;
        float retr = 0.0f;
        for (int tt = 0; tt < 3; ++tt) retr += Wr[sh_topi[g][tt]] * sh_topv[g][tt];
        qbuf[r] = tau * qbuf[r] + (1.0f - tau) * retr;
    }
    __syncthreads();

    // ---- stage 8: modulated conv weight + effective bias / scale ----------
    for (int idx = t; idx < 12288; idx += 256) {
        const int co  = idx / 192;
        const int rem = idx - co * 192;
        const int ci  = rem / 3;
        const int k   = rem - ci * 3;
        wmod[idx] = conv_weight[idx] *
                    (1.0f + qbuf[co * 64 + ci]) *
                    (1.0f + qbuf[4096 + co * 3 + k]);
    }
    if (t < 64) {
        beff[t] = bias[t] * (1.0f + qbuf[4288 + t]);
        feff[t] = 1.0f + qbuf[4352 + t];
    }
}

// ---------------------------------------------------------------------------
// Kernel 2: conv1d(K=3, pad=1) as fp32 WMMA GEMM, double-buffered LDS staging.
//   Interior chunks: one TENSOR_LOAD_TO_LDS (TDM) per chunk, issued by wave 0
//     (2-D tile: 64 rows x 136 floats, row stride 8192; right-edge padding
//     comes from the TDM's OOB-zero behavior via clipped tensor_dim0).
//   Left-boundary chunk (l0==0, once per batch sample n): per-lane
//     global_load_async_to_lds_b128 with explicit zero halo.
//   Pipeline: prefetch chunk c+1 into the other buffer while chunk c runs
//   384 V_WMMA_F32_16X16X4_F32 per wave; s_wait_{async,tensor}cnt + barrier.
// ---------------------------------------------------------------------------
#define LROW   136            // LDS row stride (floats): 2*136 % 64 == 16 -> conflict-free
#define BUFSZ  (64 * LROW)    // 8704 floats per buffer
#define LDS_BYTES (2 * BUFSZ * 4)  // 69632 B dynamic LDS (two buffers)

// per-lane async fill (only used for the l0==0 boundary chunk)
__device__ __forceinline__ void issue_tile_fill_async(
    float* __restrict__ xs, const float* __restrict__ xn,
    int bufbase, int l0, int t)
{
    #pragma unroll 1
    for (int it = 0; it < 17; ++it) {
        const int idx = t + it * 128;            // 0..2175
        const int ci  = idx / 34;
        const int seg = idx - ci * 34;
        const int l   = l0 - 4 + seg * 4;
        float* dst = &xs[bufbase + ci * LROW + seg * 4];
        if (l < 0 || l > LLEN - 4) {
            dst[0] = 0.0f; dst[1] = 0.0f; dst[2] = 0.0f; dst[3] = 0.0f;
        } else {
            unsigned           ldsAddr = (unsigned)(size_t)dst;
            unsigned long long ga      = (unsigned long long)(xn + (size_t)ci * LLEN + l);
            asm volatile("global_load_async_to_lds_b128 %0, %1, off"
                         :: "v"(ldsAddr), "v"(ga) : "memory");
        }
    }
}

// TDM fill: one instruction moves the whole 64 x 136-float tile.
// rem_dim0 = floats from tile start to end of the x row (clips right edge).
__device__ __forceinline__ void issue_tile_fill_tdm(
    const float* __restrict__ gsrc, unsigned lds_byte_addr, unsigned rem_dim0)
{
    const unsigned long long ga = (unsigned long long)gsrc;

    v4u g0;
    g0[0] = 1u;                                   // count=1, user descriptor
    g0[1] = lds_byte_addr;                        // lds_addr [63:32]
    g0[2] = (unsigned)ga;                         // global_addr lo
    g0[3] = ((unsigned)(ga >> 32) & 0x01FFFFFFu)  // global_addr [56:32]
            | (2u << 30);                         // type = 2 ("image")

    v8u g1;
    g1[0] = 0x20000u;                             // wg_mask=0, data_size=2 (4B)
    g1[1] = (rem_dim0 & 0xFFFFu) << 16;           // tensor_dim0 lo16 (bits 79:48)
    g1[2] = ((rem_dim0 >> 16) & 0xFFFFu)          // tensor_dim0 hi16
            | (64u << 16);                        // tensor_dim1 = 64 (lo16)
    g1[3] = (136u << 16);                         // tensor_dim1 hi=0 | tile_dim0=136
    g1[4] = 64u;                                  // tile_dim1=64, tile_dim2=0 (unused)
    g1[5] = 8192u;                                // tensor_dim0_stride lo32
    g1[6] = 0u;                                   // stride0 hi16, stride1 lo16
    g1[7] = 0u;                                   // stride1 hi32

    v4u g2;
    g2[0] = 1u;                                   // tensor_dim2 = 1 (benign)
    g2[1] = 1u;                                   // tensor_dim3 = 1
    g2[2] = 0u;                                   // tensor_dim2_stride lo
    g2[3] = 0u;                                   // stride hi | tile_dim3=0 (unused)

    v4u g3;
    g3[0] = 0u;                                   // tensor_dim3_stride lo
    g3[1] = 0x10000u;                             // stride hi | tensor_dim4=1 (lo16)
    g3[2] = 0u;                                   // dim4 hi | tile_dim4=0 (unused)
    g3[3] = 0u;

    asm volatile("tensor_load_to_lds %0, %1, %2, %3"
                 :: "s"(g0), "s"(g1), "s"(g2), "s"(g3)
                 : "memory");
}

__global__ __launch_bounds__(128) void conv_wmma_kernel(
    const float* __restrict__ x,
    const float* __restrict__ wmod,
    const float* __restrict__ beff,
    const float* __restrict__ feff,
    float* __restrict__ out)
{
    extern __shared__ float xs[];   // 2 * 8704 floats, set at launch

    const int t    = threadIdx.x;
    const int wave = t >> 5;
    const int lane = t & 31;
    const int half = lane >> 4;     // 0: K rows 0..1 / M,N 0..15 ; 1: K rows 2..3
    const int j    = lane & 15;
    const int n    = blockIdx.y;
    const int cb   = wave * 16;

    // A fragments: 16x4 fp32 tile per (ci-chunk, tap). lane layout:
    //   v0 = W[co=cb+j][ci0 + 2*half], v1 = W[co=cb+j][ci0 + 2*half + 1]
    v2f afrag[16][3];
    {
        const int co = cb + j;
        #pragma unroll
        for (int cc = 0; cc < 16; ++cc) {
            const int ci = cc * 4 + 2 * half;
            #pragma unroll
            for (int k = 0; k < 3; ++k) {
                afrag[cc][k].x = wmod[(co * 64 + ci) * 3 + k];
                afrag[cc][k].y = wmod[(co * 64 + ci + 1) * 3 + k];
            }
        }
    }

    // effective bias/scale for the 8 accumulator rows this lane stores
    float be[8], fe[8];
    #pragma unroll
    for (int r = 0; r < 8; ++r) {
        const int co = cb + r + 8 * half;
        be[r] = beff[co];
        fe[r] = feff[co];
    }

    const float* xn = x + (size_t)n * C_IN * LLEN;
    const int lbase = blockIdx.x * 1024;

    // fill dispatcher: window [l0-4, l0+132)
    auto fill = [&](int bufbase, int l0) {
        if (l0 == 0) {
            issue_tile_fill_async(xs, xn, bufbase, l0, t);
        } else if (wave == 0) {
            const unsigned ldsA = (unsigned)(size_t)(&xs[bufbase]);
            issue_tile_fill_tdm(xn + (l0 - 4), ldsA, (unsigned)(8196 - l0));
        }
    };

    // ---- prologue: fill buffer 0 with chunk 0 -----------------------------
    fill(0, lbase);
    asm volatile("s_wait_asynccnt 0" ::: "memory");
    __builtin_amdgcn_s_wait_tensorcnt(0);
    __syncthreads();

    int cur = 0;
    for (int c = 0; c < 8; ++c) {
        const int l0 = lbase + c * 128;

        // ---- prefetch next chunk into the other buffer (no wait yet) -----
        if (c < 7)
            fill((cur ^ 1) * BUFSZ, l0 + 128);

        const float* xb = xs + cur * BUFSZ;

        // ---- compute: acc[lt] over 48 fp32 WMMA K-steps ------------------
        v8f acc[8] = {};
        #pragma unroll
        for (int cc = 0; cc < 16; ++cc) {
            const int row0 = (cc * 4 + 2 * half) * LROW;
            #pragma unroll
            for (int k = 0; k < 3; ++k) {
                #pragma unroll
                for (int lt = 0; lt < 8; ++lt) {
                    // B 4x16 tile: v0 = x[ci0+2*half][l], v1 = x[ci0+2*half+1][l]
                    const int o = row0 + lt * 16 + j + 3 + k; // l = l0+lt*16+j+k-1
                    v2f b;
                    b.x = xb[o];
                    b.y = xb[o + LROW];
                    acc[lt] = __builtin_amdgcn_wmma_f32_16x16x4_f32(
                        false, afrag[cc][k], false, b, (short)0, acc[lt],
                        false, false);
                }
            }
        }

        // ---- epilogue: (acc + bias_eff) * f_eff, store -------------------
        #pragma unroll
        for (int lt = 0; lt < 8; ++lt) {
            #pragma unroll
            for (int r = 0; r < 8; ++r) {
                const int co = cb + r + 8 * half;
                const float v = (acc[lt][r] + be[r]) * fe[r];
                out[((size_t)n * C_OUT + co) * LLEN + l0 + lt * 16 + j] = v;
            }
        }

        // ---- ensure prefetch landed, and all waves done reading cur ------
        asm volatile("s_wait_asynccnt 0" ::: "memory");
        __builtin_amdgcn_s_wait_tensorcnt(0);
        __syncthreads();
        cur ^= 1;
    }
}

// ---------------------------------------------------------------------------
extern "C" void kernel_launch(void* const* d_in, const int* in_sizes, int n_in,
                              void* d_out, int out_size, void* d_ws, size_t ws_size,
                              hipStream_t stream) {
    const float* x           = (const float*)d_in[0];
    const float* conv_weight = (const float*)d_in[1];
    const float* bias        = (const float*)d_in[2];
    const float* grads       = (const float*)d_in[3];
    const float* q_ema       = (const float*)d_in[4];
    const float* ctrl_w      = (const float*)d_in[5];
    const float* ctrl_b      = (const float*)d_in[6];
    const float* gp_w        = (const float*)d_in[7];
    const float* gp_b        = (const float*)d_in[8];
    const float* gt_w        = (const float*)d_in[9];
    const float* gt_b        = (const float*)d_in[10];
    const float* ln_g        = (const float*)d_in[11];
    const float* ln_b        = (const float*)d_in[12];
    const float* ccin_w      = (const float*)d_in[13];
    const float* ccin_b      = (const float*)d_in[14];
    const float* ck_w        = (const float*)d_in[15];
    const float* ck_b        = (const float*)d_in[16];
    const float* cb_w        = (const float*)d_in[17];
    const float* cb_b        = (const float*)d_in[18];
    const float* cf_w        = (const float*)d_in[19];
    const float* cf_b        = (const float*)d_in[20];
    const float* tau_w       = (const float*)d_in[21];
    const float* tau_b       = (const float*)d_in[22];
    const float* mem_W       = (const float*)d_in[23];
    const float* temp_logit  = (const float*)d_in[24];

    float* ws = (float*)d_ws;

    controller_kernel<<<1, 256, 0, stream>>>(
        conv_weight, bias, grads, q_ema, ctrl_w, ctrl_b, gp_w, gp_b,
        gt_w, gt_b, ln_g, ln_b, ccin_w, ccin_b, ck_w, ck_b,
        cb_w, cb_b, cf_w, cf_b, tau_w, tau_b, mem_W, temp_logit, ws);

    conv_wmma_kernel<<<dim3(8, N_BATCH), 128, LDS_BYTES, stream>>>(
        x, ws + WS_WMOD, ws + WS_BEFF, ws + WS_FEFF, (float*)d_out);
}